// RegularizedFMNet_vectorized_27264452395495
// MI455X (gfx1250) — compile-verified
//
#include <hip/hip_runtime.h>
#include <stdint.h>

typedef __attribute__((ext_vector_type(2))) float v2f;
typedef __attribute__((ext_vector_type(8))) float v8f;

#define NB 8
#define NV 20000
#define NK 128
#define NC 256
#define LMBDA_F 100.0f
#define NCHMAX 5

#define ESTR 36   // LDS row stride (floats) for E slab: 16B-aligned rows, bank-spread
#define FSTR 68   // LDS row stride (floats) for F slab

// ---------------------------------------------------------------------------
// CDNA5 async global->LDS copy (ASYNCcnt) helpers, inline asm per ISA 10.x
// ---------------------------------------------------------------------------
__device__ __forceinline__ void async_b128(unsigned int lds_byte, unsigned long long gaddr) {
    asm volatile("global_load_async_to_lds_b128 %0, %1, off"
                 :: "v"(lds_byte), "v"(gaddr) : "memory");
}
__device__ __forceinline__ void wait_async_le6() {
    asm volatile("s_wait_asynccnt 0x6" ::: "memory");
}
__device__ __forceinline__ void wait_async_0() {
    asm volatile("s_wait_asynccnt 0x0" ::: "memory");
}

// ---------------------------------------------------------------------------
// Resolvent mask: D[b,j,i] = (f(ey_j)-f(ex_i))^2 + (g(ey_j)-g(ex_i))^2
// f(t)=t/(t^2+1), g(t)=1/(t^2+1), t=sqrt(eval/scale)   (gamma = 0.5)
// ---------------------------------------------------------------------------
__global__ __launch_bounds__(NK)
void mask_kernel(const float* __restrict__ evx, const float* __restrict__ evy,
                 float* __restrict__ D)
{
    __shared__ float red[NK];
    __shared__ float exs[NK];
    __shared__ float eys[NK];
    const int b = blockIdx.x;
    const int t = threadIdx.x;

    const float vx = evx[b * NK + t];
    const float vy = evy[b * NK + t];
    red[t] = fmaxf(vx, vy);
    __syncthreads();
    for (int s = NK / 2; s > 0; s >>= 1) {
        if (t < s) red[t] = fmaxf(red[t], red[t + s]);
        __syncthreads();
    }
    const float inv_scale = 1.0f / red[0];
    exs[t] = sqrtf(vx * inv_scale);
    eys[t] = sqrtf(vy * inv_scale);
    __syncthreads();

    const float ey = eys[t];
    const float gy = 1.0f / (ey * ey + 1.0f);
    const float fy = ey * gy;
    float* Drow = D + ((size_t)b * NK + t) * NK;
    #pragma unroll 4
    for (int i = 0; i < NK; ++i) {
        const float ex = exs[i];
        const float gx = 1.0f / (ex * ex + 1.0f);
        const float fx = ex * gx;
        const float re = fy - fx;
        const float im = gy - gx;
        Drow[i] = re * re + im * im;
    }
}

// ---------------------------------------------------------------------------
// Spectral projection partial: Pp[chunk,b] (K x C) += E[b](K x Vchunk) * F[b](Vchunk x C)
// 8-wave workgroup computes a 128(M) x 64(N) tile; V-slab (32) double-buffered
// in LDS via async global->LDS b128 copies; fragments fed from LDS; fp32 WMMA.
// ---------------------------------------------------------------------------
__global__ __launch_bounds__(256)
void proj_wmma_async_kernel(const float* __restrict__ E,   // (B,K,V)
                            const float* __restrict__ F,   // (B,V,C)
                            float* __restrict__ Pp,        // (nchunk,B,K,C)
                            int vlen)
{
    __shared__ float Esh[2][128 * ESTR];   // 2 x 18432 B
    __shared__ float Fsh[2][32 * FSTR];    // 2 x  8704 B

    const int t    = threadIdx.x;
    const int lane = t & 31;
    const int w    = t >> 5;            // wave id 0..7 -> M-tile
    const int nn   = lane & 15;
    const int half = lane >> 4;
    const int kb   = half << 1;

    const int zz    = blockIdx.z;
    const int chunk = zz / NB;
    const int b     = zz - chunk * NB;
    const int n0    = blockIdx.x << 6;
    const int v0    = chunk * vlen;
    const int slabs = vlen >> 5;        // vlen divisible by 32

    const float* Eg = E + (size_t)b * NK * NV + v0;           // + row*NV + voff
    const float* Fg = F + ((size_t)b * NV + v0) * NC + n0;    // + vrow*NC + coff

    auto issue = [&](int s, int bufi) {
        const unsigned int Ebase = (unsigned int)(uintptr_t)&Esh[bufi][0];
        const unsigned int Fbase = (unsigned int)(uintptr_t)&Fsh[bufi][0];
        const int vs = s << 5;
        #pragma unroll
        for (int i = 0; i < 4; ++i) {              // E slab: 1024 float4 chunks
            const int c   = t + (i << 8);
            const int row = c >> 3;
            const int q4  = (c & 7) << 2;
            async_b128(Ebase + (unsigned int)(row * ESTR + q4) * 4u,
                       (unsigned long long)(uintptr_t)(Eg + (size_t)row * NV + vs + q4));
        }
        #pragma unroll
        for (int i = 0; i < 2; ++i) {              // F slab: 512 float4 chunks
            const int c  = t + (i << 8);
            const int vr = c >> 4;
            const int q4 = (c & 15) << 2;
            async_b128(Fbase + (unsigned int)(vr * FSTR + q4) * 4u,
                       (unsigned long long)(uintptr_t)(Fg + (size_t)(vs + vr) * NC + q4));
        }
    };

    v8f acc0 = {}, acc1 = {}, acc2 = {}, acc3 = {};

    issue(0, 0);
    for (int s = 0; s < slabs; ++s) {
        const int cur = s & 1;
        if (s + 1 < slabs) { issue(s + 1, cur ^ 1); wait_async_le6(); }
        else               { wait_async_0(); }
        __syncthreads();   // slab 'cur' fully in LDS for all waves

        const float* Eb = &Esh[cur][(w * 16 + nn) * ESTR];
        const float* Fb = &Fsh[cur][nn];
        #pragma unroll
        for (int vv = 0; vv < 32; vv += 4) {
            v2f a = *(const v2f*)(Eb + vv + kb);
            const float* F0 = Fb + (vv + kb) * FSTR;
            const float* F1 = F0 + FSTR;
            v2f b0 = { F0[0],  F1[0]  };
            v2f b1 = { F0[16], F1[16] };
            v2f b2 = { F0[32], F1[32] };
            v2f b3 = { F0[48], F1[48] };
            acc0 = __builtin_amdgcn_wmma_f32_16x16x4_f32(false, a, false, b0, (short)0, acc0, false, false);
            acc1 = __builtin_amdgcn_wmma_f32_16x16x4_f32(false, a, false, b1, (short)0, acc1, false, false);
            acc2 = __builtin_amdgcn_wmma_f32_16x16x4_f32(false, a, false, b2, (short)0, acc2, false, false);
            acc3 = __builtin_amdgcn_wmma_f32_16x16x4_f32(false, a, false, b3, (short)0, acc3, false, false);
        }
        __syncthreads();   // all reads done before next overwrite of this buffer
    }

    float* Orow = Pp + (((size_t)chunk * NB + b) * NK + w * 16 + (half << 3)) * NC + n0 + nn;
    #pragma unroll
    for (int r = 0; r < 8; ++r) {
        Orow[(size_t)r * NC +  0] = acc0[r];
        Orow[(size_t)r * NC + 16] = acc1[r];
        Orow[(size_t)r * NC + 32] = acc2[r];
        Orow[(size_t)r * NC + 48] = acc3[r];
    }
}

// ---------------------------------------------------------------------------
// Deterministic fixed-order reduction of V-chunk partials.
// ---------------------------------------------------------------------------
__global__ __launch_bounds__(256)
void reduce_kernel(const float* __restrict__ Pp, float* __restrict__ Out,
                   int nchunk, size_t n)
{
    const size_t i = (size_t)blockIdx.x * 256 + threadIdx.x;
    float s = 0.0f;
    for (int c = 0; c < nchunk; ++c) s += Pp[(size_t)c * n + i];
    Out[i] = s;
}

// ---------------------------------------------------------------------------
// Gram: G[b,i,j] = sum_c P[b,i,c] * Q[b,j,c]   (NT-GEMM, K-dim = C = 256)
// ---------------------------------------------------------------------------
__global__ __launch_bounds__(32)
void gram_wmma_kernel(const float* __restrict__ P,  // (B,K,C)
                      const float* __restrict__ Q,  // (B,K,C)
                      float* __restrict__ G)        // (B,K,K)
{
    const int lane = threadIdx.x;
    const int nn = lane & 15;
    const int kb = (lane >> 4) << 1;
    const int b  = blockIdx.z;
    const int i0 = blockIdx.y << 4;
    const int j0 = blockIdx.x << 4;

    const float* Prow = P + ((size_t)b * NK + i0 + nn) * NC;
    const float* Qrow = Q + ((size_t)b * NK + j0 + nn) * NC;

    v8f acc = {};
    #pragma unroll 4
    for (int c = 0; c < NC; c += 4) {
        v2f a  = *(const v2f*)(Prow + c + kb);
        v2f bb = *(const v2f*)(Qrow + c + kb);
        acc = __builtin_amdgcn_wmma_f32_16x16x4_f32(false, a, false, bb, (short)0, acc, false, false);
    }

    float* Grow = G + ((size_t)b * NK + i0 + ((lane >> 4) << 3)) * NK + j0 + nn;
    #pragma unroll
    for (int r = 0; r < 8; ++r)
        Grow[(size_t)r * NK] = acc[r];
}

// ---------------------------------------------------------------------------
// Batched solve: M = AAt[b] + 100*diag(D[b,i,:]) (SPD), M x = BAt[b,i,:].
// One 128-thread workgroup per system; LDS-staged GE, 129-stride padding.
// ---------------------------------------------------------------------------
__global__ __launch_bounds__(NK)
void solve_kernel(const float* __restrict__ AAt,   // (B,K,K)
                  const float* __restrict__ BAt,   // (B,K,K)
                  const float* __restrict__ D,     // (B,K,K)
                  float* __restrict__ Out)         // (B,K,K)
{
    extern __shared__ float lds[];
    float* M   = lds;                       // NK x (NK+1)
    float* rhs = lds + NK * (NK + 1);
    float* xv  = rhs + NK;

    const int sys = blockIdx.x;             // b*NK + i
    const int b   = sys >> 7;
    const int r   = threadIdx.x;

    const float* Ab = AAt + (size_t)b * NK * NK;
    for (int idx = r; idx < NK * NK; idx += NK) {
        const int row = idx >> 7;
        const int col = idx & (NK - 1);
        M[row * (NK + 1) + col] = Ab[idx];
    }
    __syncthreads();
    M[r * (NK + 1) + r] += LMBDA_F * D[(size_t)sys * NK + r];
    rhs[r] = BAt[(size_t)sys * NK + r];
    __syncthreads();

    for (int k = 0; k < NK - 1; ++k) {
        if (r > k) {
            const float* Mk = M + k * (NK + 1);
            float*       Mr = M + r * (NK + 1);
            const float f = Mr[k] / Mk[k];
            #pragma unroll 4
            for (int j = k + 1; j < NK; ++j)
                Mr[j] -= f * Mk[j];
            rhs[r] -= f * rhs[k];
        }
        __syncthreads();
    }

    for (int k = NK - 1; k >= 0; --k) {
        if (r == k) xv[k] = rhs[k] / M[k * (NK + 1) + k];
        __syncthreads();
        if (r < k) rhs[r] -= M[r * (NK + 1) + k] * xv[k];
        __syncthreads();
    }

    Out[(size_t)sys * NK + r] = xv[r];
}

// ---------------------------------------------------------------------------
extern "C" void kernel_launch(void* const* d_in, const int* in_sizes, int n_in,
                              void* d_out, int out_size, void* d_ws, size_t ws_size,
                              hipStream_t stream) {
    const float* feat_x  = (const float*)d_in[0];   // (B,V,C)
    const float* feat_y  = (const float*)d_in[1];   // (B,V,C)
    const float* evals_x = (const float*)d_in[2];   // (B,K)
    const float* evals_y = (const float*)d_in[3];   // (B,K)
    const float* etx     = (const float*)d_in[4];   // (B,K,V)
    const float* ety     = (const float*)d_in[5];   // (B,K,V)
    float* out = (float*)d_out;                     // (B,K,K)

    const size_t nkc = (size_t)NB * NK * NC;        // 262144
    const size_t nkk = (size_t)NB * NK * NK;        // 131072

    // choose V-chunk count based on available scratch (deterministic: ws_size fixed)
    int nchunk = NCHMAX;
    {
        const size_t needed = (2 * nkc + 3 * nkk + 2 * (size_t)NCHMAX * nkc) * sizeof(float);
        if (ws_size < needed) nchunk = 1;
    }
    const int vlen = NV / nchunk;                   // 4000 or 20000, both % 32 == 0

    float* ws  = (float*)d_ws;
    float* Aw  = ws;                                 // B*K*C
    float* Bw  = Aw  + nkc;                          // B*K*C
    float* AAt = Bw  + nkc;                          // B*K*K
    float* BAt = AAt + nkk;                          // B*K*K
    float* Dm  = BAt + nkk;                          // B*K*K
    float* PpA = Dm  + nkk;                          // nchunk*B*K*C
    float* PpB = PpA + (size_t)nchunk * nkc;         // nchunk*B*K*C

    mask_kernel<<<NB, NK, 0, stream>>>(evals_x, evals_y, Dm);

    dim3 gp(NC / 64, 1, NB * nchunk);
    proj_wmma_async_kernel<<<gp, 256, 0, stream>>>(etx, feat_x, PpA, vlen);
    proj_wmma_async_kernel<<<gp, 256, 0, stream>>>(ety, feat_y, PpB, vlen);

    reduce_kernel<<<(unsigned)(nkc / 256), 256, 0, stream>>>(PpA, Aw, nchunk, nkc);
    reduce_kernel<<<(unsigned)(nkc / 256), 256, 0, stream>>>(PpB, Bw, nchunk, nkc);

    dim3 gg(NK / 16, NK / 16, NB);
    gram_wmma_kernel<<<gg, 32, 0, stream>>>(Aw, Aw, AAt);
    gram_wmma_kernel<<<gg, 32, 0, stream>>>(Bw, Aw, BAt);

    const size_t shmem = (size_t)(NK * (NK + 1) + 2 * NK) * sizeof(float);
    solve_kernel<<<NB * NK, NK, shmem, stream>>>(AAt, BAt, Dm, out);
}